// CausalLaCTSwiGLUWithSlidingWindowAttn_6390911336714
// MI455X (gfx1250) — compile-verified
//
#include <hip/hip_runtime.h>
#include <cstdint>

#define DIM      2048
#define LSEQ     8192
#define NHEAD    16
#define HD       128
#define CHUNK_T  2048
#define QKVW     6144

typedef __attribute__((ext_vector_type(2))) float v2f;
typedef __attribute__((ext_vector_type(8))) float v8f;

__constant__ float NS_A[5] = {4.0848f, 3.9505f, 3.7418f, 2.8769f, 2.8366f};
__constant__ float NS_B[5] = {-6.8946f, -6.3029f, -5.5913f, -3.1427f, -3.0525f};
__constant__ float NS_C[5] = {2.927f, 2.6377f, 2.3037f, 1.2046f, 1.2012f};

__device__ __forceinline__ v8f wmma4(v2f a, v2f b, v8f c) {
  // D = A(16x4 f32) x B(4x16 f32) + C(16x16 f32)
  return __builtin_amdgcn_wmma_f32_16x16x4_f32(false, a, false, b, (short)0, c, false, false);
}

__device__ __forceinline__ float wred(float s) {
#pragma unroll
  for (int m = 16; m > 0; m >>= 1) s += __shfl_xor(s, m, 32);
  return s;
}

__device__ __forceinline__ float dev_silu(float x) { return x / (1.f + expf(-x)); }

// ---------------------------------------------------------------------------
// CDNA5 async global->LDS copies (ASYNCcnt-tracked, bypass VGPRs).
// Low 32 bits of a flat pointer into LDS are the LDS byte address (aperture
// mapping discards the upper bits), so a generic shared pointer cast is a
// valid per-lane VDST operand.
__device__ __forceinline__ void async_copy16(void* lds_dst, const void* gsrc) {
  const uint32_t ldsa = (uint32_t)(uintptr_t)lds_dst;
  asm volatile("global_load_async_to_lds_b128 %0, %1, off"
               :: "v"(ldsa), "v"(gsrc) : "memory");
}
__device__ __forceinline__ void async_copy4(void* lds_dst, const void* gsrc) {
  const uint32_t ldsa = (uint32_t)(uintptr_t)lds_dst;
  asm volatile("global_load_async_to_lds_b32 %0, %1, off"
               :: "v"(ldsa), "v"(gsrc) : "memory");
}
__device__ __forceinline__ void async_wait0() {
  asm volatile("s_wait_asynccnt 0x0" ::: "memory");
}

// ---------------------------------------------------------------------------
// Block-cooperative LDS->LDS GEMM: C(128 x NT*16) = op(A)(128x128) @ op(B).
// 8 waves, wave w owns rows [16w,16w+16). k-outer, NT accumulators in VGPRs:
// one A-fragment load feeds NT independent wmmas.
// TA: element (m,k) read as As[k*lda+m]. TB: element (k,n) read as Bs[n*ldb+k].
template <int NT, bool TA, bool TB>
__device__ inline void gemm128t(const float* As, int lda,
                                const float* Bs, int ldb,
                                float* Cs, int ldc)
{
  const int tid = threadIdx.x, wave = tid >> 5, lane = tid & 31;
  const int m0 = wave * 16, lh = lane >> 4, lm = lane & 15;
  const int m = m0 + lm;
  v8f acc[NT];
#pragma unroll
  for (int nt = 0; nt < NT; ++nt)
#pragma unroll
    for (int i = 0; i < 8; ++i) acc[nt][i] = 0.f;
  for (int k0 = 0; k0 < 128; k0 += 4) {
    const int kb = k0 + lh * 2;
    v2f a;
    if (TA) { a.x = As[kb * lda + m]; a.y = As[(kb + 1) * lda + m]; }
    else    { a.x = As[m * lda + kb]; a.y = As[m * lda + kb + 1]; }
#pragma unroll
    for (int nt = 0; nt < NT; ++nt) {
      const int n = nt * 16 + lm;
      v2f b;
      if (TB) { b.x = Bs[n * ldb + kb]; b.y = Bs[n * ldb + kb + 1]; }
      else    { b.x = Bs[kb * ldb + n]; b.y = Bs[(kb + 1) * ldb + n]; }
      acc[nt] = wmma4(a, b, acc[nt]);
    }
  }
#pragma unroll
  for (int nt = 0; nt < NT; ++nt)
#pragma unroll
    for (int i = 0; i < 8; ++i)
      Cs[(m0 + lh * 8 + i) * ldc + nt * 16 + lm] = acc[nt][i];
}

// ---------------------------------------------------------------------------
// Streaming dw GEMM: X(128x128) = scale * sum_k A(:,k) B(k,:). Operands via
// functors (may apply per-k lr scaling), staged in LDS, acc in VGPRs.
template <class FA, class FB>
__device__ inline void gemm_dw(FA loadA, FB loadB, int K,
                               float* tileA, float* tileB, float* Xout, float scale)
{
  const int tid = threadIdx.x, wave = tid >> 5, lane = tid & 31;
  const int m0 = wave * 16, lh = lane >> 4, lm = lane & 15;
  v8f acc[8];
  for (int nt = 0; nt < 8; ++nt)
#pragma unroll
    for (int i = 0; i < 8; ++i) acc[nt][i] = 0.f;
  for (int k0 = 0; k0 < K; k0 += 32) {
    __syncthreads();
    for (int idx = tid; idx < 4096; idx += 256) {
      const int mm = idx >> 5, kk = idx & 31;
      tileA[mm * 32 + kk] = loadA(mm, k0 + kk);
      const int kr = idx >> 7, nn = idx & 127;
      tileB[kr * 128 + nn] = loadB(k0 + kr, nn);
    }
    __syncthreads();
    for (int ks = 0; ks < 32; ks += 4) {
      const int kb = ks + lh * 2;
      v2f a;
      a.x = tileA[(m0 + lm) * 32 + kb];
      a.y = tileA[(m0 + lm) * 32 + kb + 1];
#pragma unroll
      for (int nt = 0; nt < 8; ++nt) {
        v2f b;
        b.x = tileB[kb * 128 + nt * 16 + lm];
        b.y = tileB[(kb + 1) * 128 + nt * 16 + lm];
        acc[nt] = wmma4(a, b, acc[nt]);
      }
    }
  }
  __syncthreads();
  for (int nt = 0; nt < 8; ++nt)
#pragma unroll
    for (int i = 0; i < 8; ++i)
      Xout[(m0 + lh * 8 + i) * 128 + nt * 16 + lm] = acc[nt][i] * scale;
}

// ---------------------------------------------------------------------------
// Newton-Schulz (5 iter) on X(128x128 in LDS), then w += X, row-renorm to wn0.
__device__ inline void ns5_update(float* Xb, float* Ab, float* Bb, float* Tb,
                                  float* wg, const float* wn0row, float* red)
{
  const int tid = threadIdx.x, wave = tid >> 5, lane = tid & 31;
  __syncthreads();
  float s = 0.f;
  for (int idx = tid; idx < 16384; idx += 256) { float v = Xb[idx]; s += v * v; }
  s = wred(s);
  if (lane == 0) red[wave] = s;
  __syncthreads();
  if (tid == 0) {
    float t = 0.f;
    for (int i = 0; i < 8; ++i) t += red[i];
    red[8] = 1.f / (sqrtf(t) + 1e-7f);
  }
  __syncthreads();
  const float inv = red[8];
  for (int idx = tid; idx < 16384; idx += 256) Xb[idx] *= inv;
  __syncthreads();
  for (int it = 0; it < 5; ++it) {
    gemm128t<8, false, true >(Xb, 128, Xb, 128, Ab, 128); __syncthreads(); // A = X X^T
    gemm128t<8, false, false>(Ab, 128, Ab, 128, Tb, 128); __syncthreads(); // T = A A
    const float ac = NS_A[it], bc = NS_B[it], cc = NS_C[it];
    for (int idx = tid; idx < 16384; idx += 256) Bb[idx] = bc * Ab[idx] + cc * Tb[idx];
    __syncthreads();
    gemm128t<8, false, false>(Bb, 128, Xb, 128, Tb, 128); __syncthreads(); // T = Bm X
    for (int idx = tid; idx < 16384; idx += 256) Xb[idx] = ac * Xb[idx] + Tb[idx];
    __syncthreads();
  }
  for (int idx = tid; idx < 16384; idx += 256) Xb[idx] = wg[idx] + Xb[idx];
  __syncthreads();
  for (int r = wave; r < 128; r += 8) {
    float ss = 0.f;
    for (int q = lane; q < 128; q += 32) { float v = Xb[r * 128 + q]; ss += v * v; }
    ss = wred(ss);
    const float scl = wn0row[r] / (sqrtf(ss) + 1e-5f);
    for (int q = lane; q < 128; q += 32) wg[r * 128 + q] = Xb[r * 128 + q] * scl;
  }
  __threadfence();
  __syncthreads();
}

// ---------------------------------------------------------------------------
// Big dense GEMM: C(MxN) = act(A(MxK) @ B(KxN)). ACT==1 -> silu.
// Async double-buffered LDS tiles: next K-tile streams in (ASYNCcnt) under
// the current tile's wmma work.
template <int ACT>
__global__ __launch_bounds__(256) void gemm_kernel(const float* __restrict__ A,
                                                   const float* __restrict__ B,
                                                   float* __restrict__ C,
                                                   int M, int N, int K)
{
  __shared__ __align__(16) float As[2][128 * 32];
  __shared__ __align__(16) float Bs[2][32 * 128];
  const int m0 = blockIdx.y * 128, n0 = blockIdx.x * 128;
  const int tid = threadIdx.x, wave = tid >> 5, lane = tid & 31;
  const int lh = lane >> 4, lm = lane & 15;

  auto load_tile = [&](int k0, int bufi) {
    for (int g = tid; g < 1024; g += 256) {
      const int r = g >> 3, c4 = (g & 7) << 2;    // A granule: row r, 4 floats
      async_copy16(&As[bufi][r * 32 + c4], A + (size_t)(m0 + r) * K + k0 + c4);
      const int kr = g >> 5, n4 = (g & 31) << 2;  // B granule: row kr, 4 floats
      async_copy16(&Bs[bufi][kr * 128 + n4], B + (size_t)(k0 + kr) * N + n0 + n4);
    }
  };

  v8f acc[8];
  for (int nt = 0; nt < 8; ++nt)
#pragma unroll
    for (int i = 0; i < 8; ++i) acc[nt][i] = 0.f;

  load_tile(0, 0);
  async_wait0();
  __syncthreads();
  int buf = 0;
  for (int k0 = 0; k0 < K; k0 += 32) {
    if (k0 + 32 < K) load_tile(k0 + 32, buf ^ 1);   // overlap with compute
    const float* Asb = As[buf];
    const float* Bsb = Bs[buf];
    const int mrow = wave * 16 + lm;
    for (int ks = 0; ks < 32; ks += 4) {
      const int kb = ks + lh * 2;
      v2f a;
      a.x = Asb[mrow * 32 + kb];
      a.y = Asb[mrow * 32 + kb + 1];
#pragma unroll
      for (int nt = 0; nt < 8; ++nt) {
        v2f b;
        b.x = Bsb[kb * 128 + nt * 16 + lm];
        b.y = Bsb[(kb + 1) * 128 + nt * 16 + lm];
        acc[nt] = wmma4(a, b, acc[nt]);
      }
    }
    async_wait0();      // next tile fully in LDS
    __syncthreads();    // all waves done reading old tile
    buf ^= 1;
  }
  for (int nt = 0; nt < 8; ++nt)
#pragma unroll
    for (int i = 0; i < 8; ++i) {
      const int row = m0 + wave * 16 + lh * 8 + i;
      const int col = n0 + nt * 16 + lm;
      float v = acc[nt][i];
      if (ACT == 1) v = v / (1.f + expf(-v));
      C[(size_t)row * N + col] = v;
    }
}

// ---------------------------------------------------------------------------
// Per (token, head): l2-normalize q and k (128 each) in-place in qkv buffer.
__global__ void l2norm_qk(float* __restrict__ qkv)
{
  const int t = blockIdx.x, h = blockIdx.y, lane = threadIdx.x;
  float* q = qkv + (size_t)t * QKVW + h * 384;
#pragma unroll
  for (int part = 0; part < 2; ++part) {
    float* p = q + part * 128;
    float v0 = p[lane], v1 = p[lane + 32], v2 = p[lane + 64], v3 = p[lane + 96];
    float ss = v0 * v0 + v1 * v1 + v2 * v2 + v3 * v3;
    ss = wred(ss);
    const float inv = 1.f / (sqrtf(ss) + 1e-5f);
    p[lane] = v0 * inv; p[lane + 32] = v1 * inv;
    p[lane + 64] = v2 * inv; p[lane + 96] = v3 * inv;
  }
}

// ---------------------------------------------------------------------------
// lr (softplus), momentum (sigmoid), scale (silu) narrow projections.
__global__ __launch_bounds__(256) void smallproj_kernel(
    const float* __restrict__ x, const float* __restrict__ Wlr,
    const float* __restrict__ Wm, const float* __restrict__ bm,
    const float* __restrict__ Ws, const float* __restrict__ bs,
    float* __restrict__ lrb, float* __restrict__ momb, float* __restrict__ scb)
{
  __shared__ float xs[2048];
  const int t = blockIdx.x, tid = threadIdx.x, wave = tid >> 5, lane = tid & 31;
  for (int i = tid; i < 2048; i += 256) xs[i] = x[(size_t)t * 2048 + i];
  __syncthreads();
  const float blinv = 0.01f + logf(-expm1f(-0.01f));
  for (int r = 0; r < 10; ++r) {
    const int o = wave * 10 + r;
    const float* W; int nc, col;
    if (o < 48)      { W = Wlr; nc = 48; col = o; }
    else if (o < 64) { W = Wm;  nc = 16; col = o - 48; }
    else             { W = Ws;  nc = 16; col = o - 64; }
    float s = 0.f;
    for (int i = lane; i < 2048; i += 32) s += xs[i] * W[(size_t)i * nc + col];
    s = wred(s);
    if (lane == 0) {
      if (o < 48) {
        const int j = o >> 4, hh = o & 15;
        const float v = s + blinv;
        lrb[(size_t)(j * 16 + hh) * LSEQ + t] = (v > 20.f) ? v : log1pf(expf(v));
      } else if (o < 64) {
        const int hh = o - 48;
        const float v = s + bm[hh];
        momb[(size_t)hh * LSEQ + t] = 1.f / (1.f + expf(-v));
      } else {
        const int hh = o - 64;
        const float v = s + bs[hh];
        scb[(size_t)hh * LSEQ + t] = v / (1.f + expf(-v));
      }
    }
  }
}

__global__ void momean_kernel(const float* __restrict__ momb, float* __restrict__ mmean)
{
  __shared__ float red[8];
  const int h = blockIdx.x & 15, c = blockIdx.x >> 4;
  const int tid = threadIdx.x, wave = tid >> 5, lane = tid & 31;
  float s = 0.f;
  for (int i = tid; i < 2048; i += 256) s += momb[(size_t)h * LSEQ + c * 2048 + i];
  s = wred(s);
  if (lane == 0) red[wave] = s;
  __syncthreads();
  if (tid == 0) {
    float t = 0.f;
    for (int i = 0; i < 8; ++i) t += red[i];
    mmean[c * 16 + h] = t * (1.f / 2048.f);
  }
}

// ---------------------------------------------------------------------------
// Per-head TTT recurrence: one workgroup per head, weights LDS-resident
// (3 x 64 KB fast weights fit CDNA5's 320 KB WGP LDS).
__global__ __launch_bounds__(256) void ttt_kernel(
    const float* __restrict__ qkv, const float* __restrict__ lrb,
    const float* __restrict__ mmean,
    const float* __restrict__ w0in, const float* __restrict__ w1in,
    const float* __restrict__ w2in,
    float* __restrict__ wbuf, float* __restrict__ wn0,
    float* __restrict__ Gbuf, float* __restrict__ outb)
{
  extern __shared__ float sm[];
  __shared__ float red[16];
  const int h = blockIdx.x;
  const int tid = threadIdx.x, wave = tid >> 5, lane = tid & 31;

  float* W0 = sm;               // 16384 floats
  float* W1 = sm + 16384;
  float* W2 = sm + 32768;
  float* TB0 = sm + 49152;      // 4 tile buffers of 128x32
  float* TB1 = TB0 + 4096;
  float* TB2 = TB1 + 4096;
  float* TB3 = TB2 + 4096;
  // NS5 aliases (weights region reused; weights reloaded per chunk)
  float* Xb = sm;
  float* Ab = sm + 16384;
  float* Bb = sm + 32768;
  float* Tb = sm + 49152;
  float* tileA = Ab;            // streaming tiles live inside Ab region
  float* tileB = Ab + 4096;

  const int qoff = h * 384, koff = qoff + 128, voff = qoff + 256;
  float* wbp[3] = { wbuf + (size_t)h * 16384,
                    wbuf + (size_t)(16 + h) * 16384,
                    wbuf + (size_t)(32 + h) * 16384 };
  const float* winp[3] = { w0in + (size_t)h * 16384,
                           w1in + (size_t)h * 16384,
                           w2in + (size_t)h * 16384 };

  // init working weights + initial per-row norms (from inputs; deterministic)
  for (int j = 0; j < 3; ++j)
    for (int idx = tid; idx < 16384; idx += 256) wbp[j][idx] = winp[j][idx];
  for (int j = 0; j < 3; ++j)
    for (int r = wave; r < 128; r += 8) {
      float ss = 0.f;
      for (int q = lane; q < 128; q += 32) { float v = winp[j][r * 128 + q]; ss += v * v; }
      ss = wred(ss);
      if (lane == 0) wn0[(size_t)(j * 16 + h) * 128 + r] = sqrtf(ss);
    }
  __threadfence();
  __syncthreads();

  for (int c = 0; c < 4; ++c) {
    const int cb = c * CHUNK_T;
    // ---- load current weights to LDS ----
    float* WL[3] = { W0, W1, W2 };
    __syncthreads();
    for (int j = 0; j < 3; ++j)
      for (int idx = tid; idx < 16384; idx += 256) WL[j][idx] = wbp[j][idx];
    __syncthreads();

    // ---- apply: out = w1 @ (silu(w0 q) * (w2 q)) ----
    for (int t0 = 0; t0 < CHUNK_T; t0 += 32) {
      __syncthreads();
      for (int idx = tid; idx < 4096; idx += 256) {
        const int d = idx >> 5, t = idx & 31;
        async_copy4(&TB0[idx], &qkv[(size_t)(cb + t0 + t) * QKVW + qoff + d]);
      }
      async_wait0();
      __syncthreads();
      gemm128t<2, false, false>(W0, 128, TB0, 32, TB1, 32);
      gemm128t<2, false, false>(W2, 128, TB0, 32, TB2, 32);
      __syncthreads();
      for (int idx = tid; idx < 4096; idx += 256)
        TB1[idx] = dev_silu(TB1[idx]) * TB2[idx];
      __syncthreads();
      gemm128t<2, false, false>(W1, 128, TB1, 32, TB2, 32);
      __syncthreads();
      for (int idx = tid; idx < 4096; idx += 256) {
        const int d = idx >> 5, t = idx & 31;
        outb[(size_t)(cb + t0 + t) * DIM + h * HD + d] = TB2[idx];
      }
    }
    if (c == 3) break;

    // ---- gradient intermediates G0/G1/G2 (128 x CHUNK) ----
    float* G0 = Gbuf + (size_t)(0 * 16 + h) * HD * CHUNK_T;
    float* G1 = Gbuf + (size_t)(1 * 16 + h) * HD * CHUNK_T;
    float* G2 = Gbuf + (size_t)(2 * 16 + h) * HD * CHUNK_T;
    const float* lr0 = lrb + (size_t)(0 * 16 + h) * LSEQ + cb;
    const float* lr1 = lrb + (size_t)(1 * 16 + h) * LSEQ + cb;
    const float* lr2 = lrb + (size_t)(2 * 16 + h) * LSEQ + cb;

    for (int t0 = 0; t0 < CHUNK_T; t0 += 32) {
      __syncthreads();
      for (int idx = tid; idx < 4096; idx += 256) {
        const int d = idx >> 5, t = idx & 31;
        async_copy4(&TB0[idx], &qkv[(size_t)(cb + t0 + t) * QKVW + koff + d]); // k^T
        async_copy4(&TB3[idx], &qkv[(size_t)(cb + t0 + t) * QKVW + voff + d]); // v^T
      }
      async_wait0();
      __syncthreads();
      gemm128t<2, false, false>(W0, 128, TB0, 32, TB1, 32);  // gate_before
      gemm128t<2, false, false>(W2, 128, TB0, 32, TB2, 32);  // hidden_before
      __syncthreads();
      gemm128t<2, true,  false>(W1, 128, TB3, 32, TB0, 32);  // dhidden = w1^T v
      __syncthreads();
      for (int idx = tid; idx < 4096; idx += 256) {
        const int d = idx >> 5, t = idx & 31;
        const int tok = t0 + t;
        const float g = TB1[idx], hh = TB2[idx], dh = TB0[idx];
        const float sg = 1.f / (1.f + expf(-g));
        const float sil = g * sg;
        G1[(size_t)d * CHUNK_T + tok] = sil * hh * lr1[tok];        // hidden*lr1
        G2[(size_t)d * CHUNK_T + tok] = dh * sil;                   // dhidden_before
        const float dgate = dh * hh;
        G0[(size_t)d * CHUNK_T + tok] = dgate * sg * (1.f + g * (1.f - sg)); // dgate_before
      }
    }
    __syncthreads();
    __threadfence();

    // ---- dw GEMMs (K=2048) + momentum + NS5 + weight update ----
    const float mm = 1.f + mmean[c * 16 + h];
    {
      auto lA = [&](int m, int kk) { return G0[(size_t)m * CHUNK_T + kk]; };
      auto lB = [&](int kk, int n) { return qkv[(size_t)(cb + kk) * QKVW + koff + n] * lr0[kk]; };
      gemm_dw(lA, lB, CHUNK_T, tileA, tileB, Xb, mm);               // dw0
      ns5_update(Xb, Ab, Bb, Tb, wbp[0], wn0 + (size_t)(0 * 16 + h) * 128, red);
    }
    {
      auto lA = [&](int m, int kk) { return qkv[(size_t)(cb + kk) * QKVW + voff + m]; };
      auto lB = [&](int kk, int n) { return G1[(size_t)n * CHUNK_T + kk]; };
      gemm_dw(lA, lB, CHUNK_T, tileA, tileB, Xb, mm);               // dw1 = v @ (hidden*lr1)^T
      ns5_update(Xb, Ab, Bb, Tb, wbp[1], wn0 + (size_t)(1 * 16 + h) * 128, red);
    }
    {
      auto lA = [&](int m, int kk) { return G2[(size_t)m * CHUNK_T + kk]; };
      auto lB = [&](int kk, int n) { return qkv[(size_t)(cb + kk) * QKVW + koff + n] * lr2[kk]; };
      gemm_dw(lA, lB, CHUNK_T, tileA, tileB, Xb, mm);               // dw2
      ns5_update(Xb, Ab, Bb, Tb, wbp[2], wn0 + (size_t)(2 * 16 + h) * 128, red);
    }
  }
}

// ---------------------------------------------------------------------------
// Per-head LayerNorm over 128 dims + output gate scale, in-place.
__global__ void ln_scale_kernel(float* __restrict__ outb, const float* __restrict__ g,
                                const float* __restrict__ bta, const float* __restrict__ scb)
{
  __shared__ float red[8];
  const int t = blockIdx.x, h = blockIdx.y, d = threadIdx.x;
  const int wave = d >> 5, lane = d & 31;
  const float v = outb[(size_t)t * DIM + h * HD + d];
  float s = wred(v);
  if (lane == 0) red[wave] = s;
  __syncthreads();
  const float mean = (red[0] + red[1] + red[2] + red[3]) * (1.f / 128.f);
  const float cdev = v - mean;
  float cs = wred(cdev * cdev);
  if (lane == 0) red[4 + wave] = cs;
  __syncthreads();
  const float var = (red[4] + red[5] + red[6] + red[7]) * (1.f / 128.f);
  const float y = cdev * rsqrtf(var + 1e-6f) * g[d] + bta[d];
  outb[(size_t)t * DIM + h * HD + d] = y * scb[(size_t)h * LSEQ + t];
}

// ---------------------------------------------------------------------------
extern "C" void kernel_launch(void* const* d_in, const int* in_sizes, int n_in,
                              void* d_out, int out_size, void* d_ws, size_t ws_size,
                              hipStream_t stream)
{
  (void)in_sizes; (void)n_in; (void)out_size; (void)ws_size;
  const float* x     = (const float*)d_in[0];
  const float* W_qkv = (const float*)d_in[1];
  const float* W_lr  = (const float*)d_in[2];
  const float* W_o   = (const float*)d_in[3];
  const float* W_m   = (const float*)d_in[4];
  const float* b_m   = (const float*)d_in[5];
  const float* W_s   = (const float*)d_in[6];
  const float* b_s   = (const float*)d_in[7];
  const float* ln_g  = (const float*)d_in[8];
  const float* ln_b  = (const float*)d_in[9];
  const float* w0    = (const float*)d_in[10];
  const float* w1    = (const float*)d_in[11];
  const float* w2    = (const float*)d_in[12];

  char* p = (char*)d_ws;
  auto carve = [&](size_t bytes) -> void* {
    void* r = (void*)p;
    p += (bytes + 255) & ~(size_t)255;
    return r;
  };
  float* qkv   = (float*)carve((size_t)LSEQ * QKVW * 4);          // 192 MB
  float* outb  = (float*)carve((size_t)LSEQ * DIM * 4);           // 64 MB (pre/post LN)
  float* lrb   = (float*)carve((size_t)3 * NHEAD * LSEQ * 4);
  float* momb  = (float*)carve((size_t)NHEAD * LSEQ * 4);
  float* mmean = (float*)carve(48 * 4);
  float* scb   = (float*)carve((size_t)NHEAD * LSEQ * 4);
  float* wbuf  = (float*)carve((size_t)3 * NHEAD * HD * HD * 4);
  float* wn0   = (float*)carve((size_t)3 * NHEAD * HD * 4);
  float* Gbuf  = (float*)carve((size_t)3 * NHEAD * HD * CHUNK_T * 4);

  // 1. qkv = silu(x @ W_qkv)
  gemm_kernel<1><<<dim3(QKVW / 128, LSEQ / 128), 256, 0, stream>>>(x, W_qkv, qkv,
                                                                   LSEQ, QKVW, DIM);
  // 2. l2-normalize q, k per (token, head)
  l2norm_qk<<<dim3(LSEQ, NHEAD), 32, 0, stream>>>(qkv);
  // 3. lr / momentum / scale projections
  smallproj_kernel<<<LSEQ, 256, 0, stream>>>(x, W_lr, W_m, b_m, W_s, b_s, lrb, momb, scb);
  // 4. per-chunk momentum means
  momean_kernel<<<48, 256, 0, stream>>>(momb, mmean);
  // 5. per-head TTT recurrence (256 KB dynamic LDS per workgroup)
  hipFuncSetAttribute((const void*)ttt_kernel,
                      hipFuncAttributeMaxDynamicSharedMemorySize, 262144);
  ttt_kernel<<<NHEAD, 256, 262144, stream>>>(qkv, lrb, mmean, w0, w1, w2,
                                             wbuf, wn0, Gbuf, outb);
  // 6. LayerNorm + gate scale (in place)
  ln_scale_kernel<<<dim3(LSEQ, NHEAD), 128, 0, stream>>>(outb, ln_g, ln_b, scb);
  // 7. final projection
  gemm_kernel<0><<<dim3(DIM / 128, LSEQ / 128), 256, 0, stream>>>(outb, W_o,
                                                                  (float*)d_out,
                                                                  LSEQ, DIM, DIM);
}